// SelfAttention3D_32229434589920
// MI455X (gfx1250) — compile-verified
//
#include <hip/hip_runtime.h>
#include <math.h>

typedef float v2f __attribute__((ext_vector_type(2)));
typedef float v8f __attribute__((ext_vector_type(8)));

#define NB   2
#define CCH  16
#define VOX  512    // D*H*W
#define EDIM 8192   // C*VOX

// ---------------------------------------------------------------------------
// Kernel 1: 1x1x1 conv (pointwise channel mix) via V_WMMA_F32_16X16X4_F32.
// One wave computes a 16(c_out) x 16(voxel) tile: D = W(16x16) * X(16x16) + b
// A 16x4 layout: lane(r,half) holds A[r][2*half], A[r][2*half+1]
// B 4x16 layout: lane(r,half) holds B[2*half][r], B[2*half+1][r]
// C/D 16x16:     vgpr v, lane(r,half) -> element (M = v+8*half, N = r)
// ---------------------------------------------------------------------------
__device__ __forceinline__ void conv_tile(const float* __restrict__ data,
                                          const float* __restrict__ Wm,
                                          const float* __restrict__ bm,
                                          float* __restrict__ outbuf,
                                          int n, int pbase, int r, int half) {
  v8f c = {};
#pragma unroll
  for (int kk = 0; kk < 4; ++kk) {
    const int k0 = kk * 4 + 2 * half;
    v2f a, b;
    a.x = Wm[r * CCH + k0];
    a.y = Wm[r * CCH + k0 + 1];
    b.x = data[n * EDIM + (k0    ) * VOX + pbase + r];
    b.y = data[n * EDIM + (k0 + 1) * VOX + pbase + r];
    c = __builtin_amdgcn_wmma_f32_16x16x4_f32(false, a, false, b,
                                              (short)0, c, false, false);
  }
#pragma unroll
  for (int v = 0; v < 8; ++v) {
    const int m = v + 8 * half;  // output channel
    outbuf[n * EDIM + m * VOX + pbase + r] = c[v] + bm[m];
  }
}

__global__ void kqv_kernel(const float* __restrict__ data,
                           const float* __restrict__ Wk, const float* __restrict__ bk,
                           const float* __restrict__ Wq, const float* __restrict__ bq,
                           const float* __restrict__ Wv, const float* __restrict__ bv,
                           float* __restrict__ kb, float* __restrict__ qb,
                           float* __restrict__ vb) {
  const int wave  = (int)(blockIdx.x * blockDim.x + threadIdx.x) >> 5;
  const int lane  = (int)threadIdx.x & 31;
  const int r     = lane & 15;
  const int half  = lane >> 4;
  const int n     = wave >> 5;          // 32 voxel-tiles per batch
  const int pbase = (wave & 31) << 4;
  conv_tile(data, Wk, bk, kb, n, pbase, r, half);
  conv_tile(data, Wq, bq, qb, n, pbase, r, half);
  conv_tile(data, Wv, bv, vb, n, pbase, r, half);
}

// ---------------------------------------------------------------------------
// Kernel 2: per-batch min/max of k. Rank-1 softmax max trick:
// m_j = max(q_j*kmax, q_j*kmin), so no E x E max pass is needed.
// ---------------------------------------------------------------------------
__global__ void kminmax_kernel(const float* __restrict__ kb,
                               float* __restrict__ kmm) {
  __shared__ float smax[256];
  __shared__ float smin[256];
  const int n = blockIdx.x;
  const int t = threadIdx.x;
  float mx = -INFINITY, mn = INFINITY;
  for (int i = t; i < EDIM; i += 256) {
    const float v = kb[n * EDIM + i];
    mx = fmaxf(mx, v);
    mn = fminf(mn, v);
  }
  smax[t] = mx; smin[t] = mn;
  __syncthreads();
  for (int s = 128; s > 0; s >>= 1) {
    if (t < s) {
      smax[t] = fmaxf(smax[t], smax[t + s]);
      smin[t] = fminf(smin[t], smin[t + s]);
    }
    __syncthreads();
  }
  if (t == 0) { kmm[n * 2] = smax[0]; kmm[n * 2 + 1] = smin[0]; }
}

// ---------------------------------------------------------------------------
// Kernel 3: attention.
// Block = 64 output columns (j) x 4 i-segments of 2048 -> 256 blocks,
// 2048 waves total (4x the previous occupancy). Each segment stages k/v in
// 512-element LDS tiles; (s,w) partials reduced across segments via LDS.
// out_att[j] = sum_i exp(k_i*q_j - m_j)*v_i / sum_i exp(k_i*q_j - m_j)
// ---------------------------------------------------------------------------
__global__ void attn_kernel(const float* __restrict__ kb,
                            const float* __restrict__ qb,
                            const float* __restrict__ vb,
                            const float* __restrict__ kmm,
                            float* __restrict__ att) {
  __shared__ float sk[4][512];
  __shared__ float sv[4][512];
  __shared__ float ps[4][64];
  __shared__ float pw[4][64];

  const int t   = (int)threadIdx.x;
  const int seg = t >> 6;                 // 0..3 : i-segment of 2048
  const int jj  = t & 63;                 // 0..63: column within block
  const int n   = (int)blockIdx.x >> 7;   // 128 blocks per batch
  const int j   = (((int)blockIdx.x & 127) << 6) + jj;

  const float qj = qb[n * EDIM + j];
  const float m  = fmaxf(qj * kmm[n * 2], qj * kmm[n * 2 + 1]);

  float s = 0.f, w = 0.f;
#pragma unroll 1
  for (int tile = 0; tile < 4; ++tile) {  // 4 x 512 = segment of 2048
    const int base = seg * 2048 + tile * 512;
    for (int i = jj; i < 512; i += 64) {
      sk[seg][i] = kb[n * EDIM + base + i];
      sv[seg][i] = vb[n * EDIM + base + i];
    }
    __syncthreads();
#pragma unroll 4
    for (int i = 0; i < 512; ++i) {
      const float e = __expf(fmaf(sk[seg][i], qj, -m));
      s += e;
      w = fmaf(e, sv[seg][i], w);
    }
    __syncthreads();
  }
  ps[seg][jj] = s;
  pw[seg][jj] = w;
  __syncthreads();
  if (seg == 0) {
    const float S = ps[0][jj] + ps[1][jj] + ps[2][jj] + ps[3][jj];
    const float W = pw[0][jj] + pw[1][jj] + pw[2][jj] + pw[3][jj];
    att[n * EDIM + j] = W / S;
  }
}

// ---------------------------------------------------------------------------
// Kernel 4: projection out = att(2x8192) @ Wp^T + bp via f32 WMMA 16x16x4.
// Bandwidth-bound stage: streams Wp (268 MB fp32) exactly once.
// Block = one 16-column tile, 8 waves split K (8 x 1024); partial C tiles
// reduced through LDS. 512 blocks x 8 waves = 4096 waves for latency hiding.
// ---------------------------------------------------------------------------
__global__ void proj_kernel(const float* __restrict__ att,
                            const float* __restrict__ Wp,
                            const float* __restrict__ bp,
                            float* __restrict__ out) {
  __shared__ float part[8][2][16];

  const int w    = (int)threadIdx.x >> 5;   // wave in block -> K segment
  const int lane = (int)threadIdx.x & 31;
  const int r    = lane & 15;
  const int half = lane >> 4;
  const int colbase = (int)blockIdx.x << 4;
  const int k0   = w * (EDIM / 8);

  // B[k][col] = Wp[(colbase+col)*E + k0+kbv + k]; lane(r,half) streams a
  // contiguous float2 from its private row segment of Wp.
  const float* __restrict__ wrow =
      Wp + (size_t)(colbase + r) * EDIM + k0 + 2 * half;

  v8f c = {};
#pragma unroll 8
  for (int kbv = 0; kbv < EDIM / 8; kbv += 4) {
    v2f a, b;
    const float2 bw = *(const float2*)(wrow + kbv);
    b.x = bw.x; b.y = bw.y;
    if (r < 2) {  // rows 0,1 = attention output; rows 2..15 zero padding
      const float2 aw =
          *(const float2*)(att + (size_t)r * EDIM + k0 + kbv + 2 * half);
      a.x = aw.x; a.y = aw.y;
    } else {
      a.x = 0.f; a.y = 0.f;
    }
    c = __builtin_amdgcn_wmma_f32_16x16x4_f32(false, a, false, b,
                                              (short)0, c, false, false);
  }
  // D rows M=0 (vgpr0) and M=1 (vgpr1) live in lanes 0-15 (half==0)
  if (half == 0) {
    part[w][0][r] = c[0];
    part[w][1][r] = c[1];
  }
  __syncthreads();
  if (threadIdx.x < 32) {
    const int row = (int)threadIdx.x >> 4;   // 0 or 1 (batch)
    const int cc  = (int)threadIdx.x & 15;   // column within tile
    float acc = bp[colbase + cc];
#pragma unroll
    for (int ww = 0; ww < 8; ++ww) acc += part[ww][row][cc];
    out[row * EDIM + colbase + cc] = acc;
  }
}

// ---------------------------------------------------------------------------
extern "C" void kernel_launch(void* const* d_in, const int* in_sizes, int n_in,
                              void* d_out, int out_size, void* d_ws, size_t ws_size,
                              hipStream_t stream) {
  const float* data = (const float*)d_in[0];
  const float* Wk   = (const float*)d_in[1];
  const float* bk   = (const float*)d_in[2];
  const float* Wq   = (const float*)d_in[3];
  const float* bq   = (const float*)d_in[4];
  const float* Wv   = (const float*)d_in[5];
  const float* bv   = (const float*)d_in[6];
  const float* Wp   = (const float*)d_in[7];
  const float* bp   = (const float*)d_in[8];

  float* ws   = (float*)d_ws;
  float* kb   = ws;                 // NB*EDIM = 16384 floats each
  float* qb   = ws + 16384;
  float* vb   = ws + 32768;
  float* attb = ws + 49152;
  float* kmm  = ws + 65536;         // 4 floats: [n*2]=kmax, [n*2+1]=kmin

  // 64 waves: (2 batches) x (32 voxel tiles of 16)
  kqv_kernel<<<8, 256, 0, stream>>>(data, Wk, bk, Wq, bq, Wv, bv, kb, qb, vb);
  kminmax_kernel<<<2, 256, 0, stream>>>(kb, kmm);
  // 256 blocks: (2 batches) x (128 tiles of 64 columns), 4-way K split each
  attn_kernel<<<256, 256, 0, stream>>>(kb, qb, vb, kmm, attb);
  // 512 blocks: one 16-column tile each, 8-way K split
  proj_kernel<<<512, 256, 0, stream>>>(attb, Wp, bp, (float*)d_out);
}